// MendGraph_9311489097946
// MI455X (gfx1250) — compile-verified
//
#include <hip/hip_runtime.h>
#include <hip/hip_bf16.h>
#include <stdint.h>

// ---------------------------------------------------------------------------
// CDNA5 async global<->LDS DMA helpers (gfx1250, ASYNCcnt-tracked).
// Per-lane b128: each lane moves 16B; a wave moves 512B per instruction.
// ---------------------------------------------------------------------------
__device__ __forceinline__ void async_load_b128(uint32_t lds_addr, const void* gaddr) {
    unsigned long long ga = (unsigned long long)gaddr;
    asm volatile("global_load_async_to_lds_b128 %0, %1, off"
                 :: "v"(lds_addr), "v"(ga) : "memory");
}
__device__ __forceinline__ void async_store_b128(void* gaddr, uint32_t lds_addr) {
    unsigned long long ga = (unsigned long long)gaddr;
    asm volatile("global_store_async_from_lds_b128 %0, %1, off"
                 :: "v"(ga), "v"(lds_addr) : "memory");
}
__device__ __forceinline__ void wait_async0() {
    asm volatile("s_wait_asynccnt 0" ::: "memory");
}
__device__ __forceinline__ uint32_t lds_off(const void* p) {
    // Generic LDS address is {SHARED_BASE, offset}; low 32 bits = LDS byte offset.
    return (uint32_t)(uintptr_t)p;
}
__device__ __forceinline__ int clipk(int v, int K) {
    return v < 0 ? 0 : (v > K ? K : v);
}

// ---------------------------------------------------------------------------
// Kernel 1: per-block exclusive scan of k_i = clip(degree_i, 0, K).
// ---------------------------------------------------------------------------
__global__ void mg_scan_blocks(const int* __restrict__ degree, int N, int K,
                               int* __restrict__ excl, int* __restrict__ partials) {
    __shared__ int s[256];
    const int t = threadIdx.x;
    const int i = blockIdx.x * 256 + t;
    int v = 0;
    if (i < N) v = clipk(degree[i], K);
    s[t] = v;
    __syncthreads();
    for (int off = 1; off < 256; off <<= 1) {
        int add = (t >= off) ? s[t - off] : 0;
        __syncthreads();
        s[t] += add;
        __syncthreads();
    }
    if (i < N) excl[i] = s[t] - v;
    if (t == 255) partials[blockIdx.x] = s[255];
}

// ---------------------------------------------------------------------------
// Kernel 2: single-block exclusive scan of the block partials (NB <= 512).
// ---------------------------------------------------------------------------
__global__ void mg_scan_partials(int* __restrict__ partials, int NB) {
    __shared__ int s[512];
    const int t = threadIdx.x;
    int v = (t < NB) ? partials[t] : 0;
    s[t] = v;
    __syncthreads();
    for (int off = 1; off < 512; off <<= 1) {
        int add = (t >= off) ? s[t - off] : 0;
        __syncthreads();
        s[t] += add;
        __syncthreads();
    }
    if (t < NB) partials[t] = s[t] - v;
}

// ---------------------------------------------------------------------------
// Kernel 3: copy x -> out rows [0,N) via async DMA, batch-pipelined.
// 8 x 512B buffers per wave; 4-chunk batches; one s_wait_asynccnt 0 per batch.
// The wait in batch n+1 retires batch n's stores before batch n+2 reuses bufs.
// ---------------------------------------------------------------------------
__global__ void mg_copy_x_async(const float4* __restrict__ src,
                                float4* __restrict__ dst, long nvec4) {
    __shared__ float4 stage[8][8][32];           // 8 waves x 8 bufs x 512B = 32KB
    const int lane = threadIdx.x & 31;
    const int wib  = threadIdx.x >> 5;
    const long wave   = (long)blockIdx.x * 8 + wib;
    const long nwaves = (long)gridDim.x * 8;
    const long nchunks = nvec4 >> 5;             // 32-float4 (512B) chunks
    const long ngroups = (nchunks + 3) >> 2;     // 4-chunk (2KB) batches

    uint32_t lds[8];
#pragma unroll
    for (int b = 0; b < 8; ++b) lds[b] = lds_off(&stage[wib][b][lane]);

    int bsel = 0;
    for (long g = wave; g < ngroups; g += nwaves) {
        const long c0 = g << 2;
#pragma unroll
        for (int t = 0; t < 4; ++t) {
            const long c = c0 + t;
            if (c < nchunks) async_load_b128(lds[bsel + t], src + (c << 5) + lane);
        }
        wait_async0();                           // loads done + prev batch stores done
#pragma unroll
        for (int t = 0; t < 4; ++t) {
            const long c = c0 + t;
            if (c < nchunks) async_store_b128(dst + (c << 5) + lane, lds[bsel + t]);
        }
        bsel ^= 4;
    }
    // generic tail (empty for N*F = 12.8M floats)
    for (long i = (nchunks << 5) + (wave << 5) + lane; i < nvec4; i += nwaves << 5)
        dst[i] = src[i];
}

// ---------------------------------------------------------------------------
// Kernel 4: one wave per node; gather its k_i generated rows into
// out rows [N+excl_i, N+excl_i+k_i). All <=K row-loads issued, ONE wait,
// then all stores (s_endpgm's implicit wait-idle covers the stores).
// ---------------------------------------------------------------------------
#define MG_KMAX 6
__global__ void mg_emit_feats(const float* __restrict__ gf,
                              const int* __restrict__ degree,
                              const int* __restrict__ excl,
                              const int* __restrict__ partials,
                              float* __restrict__ outF, int N, int K, int F) {
    __shared__ float stage[8][MG_KMAX][128];     // 8 waves x 6 rows x 512B = 24KB
    const int tid  = blockIdx.x * blockDim.x + threadIdx.x;
    const int lane = threadIdx.x & 31;
    const int wib  = threadIdx.x >> 5;
    const int node = tid >> 5;
    if (node >= N) return;
    int k = clipk(degree[node], K);
    if (k == 0) return;
    const long e = (long)excl[node] + (long)partials[node >> 8];
    const int chunks = F >> 2;                   // float4s per row (32 for F=128)
    if (F <= 128 && (F & 3) == 0 && K <= MG_KMAX) {
        for (int j = 0; j < k; ++j) {
            const float* srcRow = gf + ((long)node * K + j) * (long)F;
            for (int c = lane; c < chunks; c += 32)
                async_load_b128(lds_off(&stage[wib][j][c << 2]), srcRow + (c << 2));
        }
        wait_async0();
        for (int j = 0; j < k; ++j) {
            float* dstRow = outF + ((long)N + e + j) * (long)F;
            for (int c = lane; c < chunks; c += 32)
                async_store_b128(dstRow + (c << 2), lds_off(&stage[wib][j][c << 2]));
        }
    } else {                                     // generic fallback
        for (int j = 0; j < k; ++j) {
            const float* srcRow = gf   + ((long)node * K + j) * (long)F;
            float*       dstRow = outF + ((long)N + e + j)   * (long)F;
            for (int c = lane; c < F; c += 32) dstRow[c] = srcRow[c];
        }
    }
}

// ---------------------------------------------------------------------------
// Kernel 5: stream original edges (int32 -> float) into [2, E+2*total] rows.
// Scalar coalesced (row-1 base not guaranteed 16B-aligned for vector stores).
// ---------------------------------------------------------------------------
__global__ void mg_copy_edges(const int* __restrict__ edge, float* __restrict__ outE,
                              long E, long ERow) {
    const long n = 2 * E;
    for (long i = (long)blockIdx.x * blockDim.x + threadIdx.x; i < n;
         i += (long)gridDim.x * blockDim.x) {
        const long row = (i >= E) ? 1 : 0;
        const long col = i - row * E;
        outE[row * ERow + col] = (float)edge[i];
    }
}

// ---------------------------------------------------------------------------
// Kernel 6: one thread per node; interleaved new edges
// row0: [src, new], row1: [new, src] at columns E + 2t, t = excl_i + j.
// ---------------------------------------------------------------------------
__global__ void mg_emit_new_edges(const int* __restrict__ degree,
                                  const int* __restrict__ excl,
                                  const int* __restrict__ partials,
                                  float* __restrict__ outE,
                                  int N, int K, long E, long ERow) {
    const int i = blockIdx.x * blockDim.x + threadIdx.x;
    if (i >= N) return;
    int k = clipk(degree[i], K);
    if (k == 0) return;
    const long e = (long)excl[i] + (long)partials[i >> 8];
    float* r0 = outE;
    float* r1 = outE + ERow;
    const float sid = (float)i;
    for (int j = 0; j < k; ++j) {
        const long t    = e + j;
        const float nid = (float)((long)N + t);
        r0[E + 2 * t]     = sid;
        r0[E + 2 * t + 1] = nid;
        r1[E + 2 * t]     = nid;
        r1[E + 2 * t + 1] = sid;
    }
}

// ---------------------------------------------------------------------------
// Host launcher. total derived from out_size (deterministic, capture-safe):
//   out_size = (N+total)*F + 2*(E + 2*total) => total = (out_size - N*F - 2E)/(F+4)
// ---------------------------------------------------------------------------
extern "C" void kernel_launch(void* const* d_in, const int* in_sizes, int n_in,
                              void* d_out, int out_size, void* d_ws, size_t ws_size,
                              hipStream_t stream) {
    const float* x      = (const float*)d_in[0];
    const int*   edge   = (const int*)d_in[1];
    const int*   degree = (const int*)d_in[2];
    const float* gf     = (const float*)d_in[3];

    const long N = in_sizes[2];
    const long F = in_sizes[0] / N;
    const long E = in_sizes[1] / 2;
    const long K = in_sizes[3] / (N * F);
    const long total = ((long)out_size - N * F - 2 * E) / (F + 4);
    const long ERow  = E + 2 * total;

    float* outF = (float*)d_out;
    float* outE = outF + (N + total) * F;

    int* excl     = (int*)d_ws;
    int* partials = excl + N;
    const int NB  = (int)((N + 255) / 256);

    // 1-2: exclusive scan of clipped degrees -> per-node placement offsets
    mg_scan_blocks<<<NB, 256, 0, stream>>>(degree, (int)N, (int)K, excl, partials);
    mg_scan_partials<<<1, 512, 0, stream>>>(partials, NB);

    // 3: x copy through the batched async global<->LDS DMA path
    mg_copy_x_async<<<512, 256, 0, stream>>>((const float4*)x, (float4*)outF,
                                             (N * F) / 4);

    // 4: gather generated rows (wave per node), batched async DMA
    {
        const long blocks = (N + 7) / 8;
        mg_emit_feats<<<(int)blocks, 256, 0, stream>>>(gf, degree, excl, partials,
                                                       outF, (int)N, (int)K, (int)F);
    }

    // 5: int->float stream of original edges
    mg_copy_edges<<<4096, 256, 0, stream>>>(edge, outE, E, ERow);

    // 6: interleaved new edges
    mg_emit_new_edges<<<(int)((N + 255) / 256), 256, 0, stream>>>(
        degree, excl, partials, outE, (int)N, (int)K, E, ERow);
}